// PoseEstimator_5325759447781
// MI455X (gfx1250) — compile-verified
//
#include <hip/hip_runtime.h>
#include <math.h>

// ---------------------------------------------------------------------------
// Chamfer distance (bidirectional, squared, mean) for 64 batches of 2048x3
// point clouds on gfx1250. Pairwise -2*x.y term computed with
// V_WMMA_F32_16X16X4_F32 (K=4 slot folds the |y|^2 constant).
// Two row-strips per wave share each B-operand LDS load (2 WMMA / ds_load).
// ---------------------------------------------------------------------------

typedef float v2f __attribute__((ext_vector_type(2)));
typedef float v8f __attribute__((ext_vector_type(8)));

#define RDIM 4
#define IDIM 16
#define NPTS 2048
#define NBATCH (RDIM * IDIM)               // 64
#define ROWGROUPS 8                        // blocks per batch (grid.y)
#define ROWS_PER_BLOCK (NPTS / ROWGROUPS)  // 256
#define THREADS 256
#define WAVES (THREADS / 32)               // 8 wave32s
#define COLTILES (NPTS / 16)               // 128

#define BIGF 3.0e38f

// ---------------------------------------------------------------------------
// Kernel 1: init workspace (col-min array to +inf, row-sum array to 0)
// ---------------------------------------------------------------------------
__global__ void chamfer_init_kernel(float* __restrict__ colmin,
                                    float* __restrict__ rowsum) {
  int idx = blockIdx.x * blockDim.x + threadIdx.x;
  if (idx < NBATCH * NPTS) colmin[idx] = BIGF;
  if (idx < NBATCH * ROWGROUPS) rowsum[idx] = 0.0f;
}

// ---------------------------------------------------------------------------
// Kernel 2: main compute. grid = (NBATCH, ROWGROUPS), block = 256 threads.
// ---------------------------------------------------------------------------
__global__ __launch_bounds__(THREADS)
void chamfer_wmma_kernel(const float* __restrict__ src,    // [R,I,N,3]
                         const float* __restrict__ tgt,    // [I,N,3]
                         const float* __restrict__ rot,    // [R,I,3]
                         const float* __restrict__ trans,  // [R,I,3]
                         const float* __restrict__ scale,  // [R,I]
                         float* __restrict__ colmin_ws,    // [NBATCH,NPTS]
                         float* __restrict__ rowsum_ws) {  // [NBATCH,ROWGROUPS]
  __shared__ float yB[NPTS * 4];            // B-format: {-2y0,-2y1,-2y2,|y|^2}
  __shared__ float xA[ROWS_PER_BLOCK * 4];  // A-format: {x0,x1,x2,1}
  __shared__ float x2s[ROWS_PER_BLOCK];     // |x|^2 per row
  __shared__ float cmin[NPTS];              // per-block column mins
  __shared__ float wsum[WAVES];             // per-wave row-min sums

  const int b = blockIdx.x;  // batch (r*IDIM + i)
  const int g = blockIdx.y;  // row group
  const int i = b % IDIM;
  const int tid = threadIdx.x;

  // ---- rotation matrix R = Rx(ax) @ Ry(ay) @ Rz(az), redundant per thread
  const float ax = rot[b * 3 + 0], ay = rot[b * 3 + 1], az = rot[b * 3 + 2];
  const float cx = cosf(ax), sx = sinf(ax);
  const float cy = cosf(ay), sy = sinf(ay);
  const float cz = cosf(az), sz = sinf(az);
  const float R00 = cy * cz,                 R01 = -cy * sz,                 R02 = sy;
  const float R10 = sx * sy * cz + cx * sz,  R11 = -sx * sy * sz + cx * cz,  R12 = -sx * cy;
  const float R20 = -cx * sy * cz + sx * sz, R21 = cx * sy * sz + sx * cz,   R22 = cx * cy;
  const float t0 = trans[b * 3 + 0], t1 = trans[b * 3 + 1], t2 = trans[b * 3 + 2];
  const float sc = scale[b];

  // ---- phase 1: build operands in LDS
  for (int m = tid; m < NPTS; m += THREADS) {
    const float* p = src + ((size_t)b * NPTS + m) * 3;
    const float p0 = p[0], p1 = p[1], p2 = p[2];
    const float ya = sc * (R00 * p0 + R01 * p1 + R02 * p2 + t0);
    const float yb = sc * (R10 * p0 + R11 * p1 + R12 * p2 + t1);
    const float yc = sc * (R20 * p0 + R21 * p1 + R22 * p2 + t2);
    yB[m * 4 + 0] = -2.0f * ya;
    yB[m * 4 + 1] = -2.0f * yb;
    yB[m * 4 + 2] = -2.0f * yc;
    yB[m * 4 + 3] = ya * ya + yb * yb + yc * yc;
  }
  for (int n = tid; n < ROWS_PER_BLOCK; n += THREADS) {
    const int gn = g * ROWS_PER_BLOCK + n;
    const float* q = tgt + ((size_t)i * NPTS + gn) * 3;
    const float q0 = q[0], q1 = q[1], q2 = q[2];
    xA[n * 4 + 0] = q0;
    xA[n * 4 + 1] = q1;
    xA[n * 4 + 2] = q2;
    xA[n * 4 + 3] = 1.0f;
    x2s[n] = q0 * q0 + q1 * q1 + q2 * q2;
  }
  for (int m = tid; m < NPTS; m += THREADS) cmin[m] = BIGF;
  __syncthreads();

  // ---- phase 2: WMMA sweep (two row strips per wave, shared B loads)
  const int lane = threadIdx.x & 31;
  const int wave = threadIdx.x >> 5;
  const int half = lane >> 4;  // 0: lanes 0-15, 1: lanes 16-31
  const int l16 = lane & 15;

  const int row0 = wave * 32;       // strip 0 base (local rows)
  const int row1 = wave * 32 + 16;  // strip 1 base

  // A operands (16x4 f32): lanes 0-15 carry K=0,1 of row l16; lanes 16-31 K=2,3.
  const v2f a0 = *(const v2f*)&xA[(row0 + l16) * 4 + half * 2];
  const v2f a1 = *(const v2f*)&xA[(row1 + l16) * 4 + half * 2];

  // |x|^2 per (VGPR, half): D VGPR v <-> row v (lanes 0-15) / row v+8 (16-31).
  float x2v0[8], x2v1[8];
#pragma unroll
  for (int v = 0; v < 8; ++v) {
    x2v0[v] = x2s[row0 + half * 8 + v];
    x2v1[v] = x2s[row1 + half * 8 + v];
  }

  float racc0[8], racc1[8];
#pragma unroll
  for (int v = 0; v < 8; ++v) { racc0[v] = BIGF; racc1[v] = BIGF; }

  // B operand lane address: column = t*16 + l16, comps {0,1} or {2,3} by half.
  const float* ybl = &yB[l16 * 4 + half * 2];

  v2f bb = *(const v2f*)ybl;  // prefetch tile 0
#pragma unroll 2
  for (int t = 0; t < COLTILES; ++t) {
    const v2f cur = bb;
    const int tn = (t + 1) & (COLTILES - 1);  // masked: avoids tail branch
    bb = *(const v2f*)(ybl + tn * 64);        // prefetch next tile

    // D[n][m] = -2 * x_n . y_m + |y_m|^2
    v8f d0 = __builtin_amdgcn_wmma_f32_16x16x4_f32(
        false, a0, false, cur, (short)0,
        (v8f){0.f, 0.f, 0.f, 0.f, 0.f, 0.f, 0.f, 0.f}, false, false);
    v8f d1 = __builtin_amdgcn_wmma_f32_16x16x4_f32(
        false, a1, false, cur, (short)0,
        (v8f){0.f, 0.f, 0.f, 0.f, 0.f, 0.f, 0.f, 0.f}, false, false);

    // running row-min (|x|^2 added after the final reduction)
#pragma unroll
    for (int v = 0; v < 8; ++v) {
      racc0[v] = fminf(racc0[v], d0[v]);
      racc1[v] = fminf(racc1[v], d1[v]);
    }

    // column min over this tile's 32 rows (full d^2 = D + |x|^2).
    // Each half reduces its own 8 rows; the LDS atomic merges halves.
    float m0 = fminf(fminf(d0[0] + x2v0[0], d0[1] + x2v0[1]),
                     fminf(d0[2] + x2v0[2], d0[3] + x2v0[3]));
    float m1 = fminf(fminf(d0[4] + x2v0[4], d0[5] + x2v0[5]),
                     fminf(d0[6] + x2v0[6], d0[7] + x2v0[7]));
    float m2 = fminf(fminf(d1[0] + x2v1[0], d1[1] + x2v1[1]),
                     fminf(d1[2] + x2v1[2], d1[3] + x2v1[3]));
    float m3 = fminf(fminf(d1[4] + x2v1[4], d1[5] + x2v1[5]),
                     fminf(d1[6] + x2v1[6], d1[7] + x2v1[7]));
    const float colc = fminf(fminf(m0, m1), fminf(m2, m3));
    atomicMin(&cmin[t * 16 + l16], colc);  // all 32 lanes; ds_min_num_f32
  }

  // ---- row-min reduction across the 16 columns held in each lane half
  float wave_rowsum = 0.0f;
#pragma unroll
  for (int v = 0; v < 8; ++v) {
    float r = racc0[v];
    r = fminf(r, __shfl_xor(r, 1, 32));
    r = fminf(r, __shfl_xor(r, 2, 32));
    r = fminf(r, __shfl_xor(r, 4, 32));
    r = fminf(r, __shfl_xor(r, 8, 32));
    wave_rowsum += r + x2v0[v];
    float q = racc1[v];
    q = fminf(q, __shfl_xor(q, 1, 32));
    q = fminf(q, __shfl_xor(q, 2, 32));
    q = fminf(q, __shfl_xor(q, 4, 32));
    q = fminf(q, __shfl_xor(q, 8, 32));
    wave_rowsum += q + x2v1[v];
  }
  // lanes 0-15 hold sums for rows [base..base+7], lanes 16-31 rows [+8..+15]
  wave_rowsum += __shfl_xor(wave_rowsum, 16, 32);
  if (lane == 0) wsum[wave] = wave_rowsum;
  __syncthreads();

  // ---- phase 3: export per-block results
  if (tid == 0) {
    float bs = 0.0f;
#pragma unroll
    for (int w = 0; w < WAVES; ++w) bs += wsum[w];
    rowsum_ws[b * ROWGROUPS + g] = bs;
  }
  for (int m = tid; m < NPTS; m += THREADS) {
    atomicMin(&colmin_ws[(size_t)b * NPTS + m], cmin[m]);  // global float min
  }
}

// ---------------------------------------------------------------------------
// Kernel 3: finalize. out[b] = mean(rowmin) + mean(colmin). Deterministic
// fixed-order sums (no float atomicAdd anywhere).
// ---------------------------------------------------------------------------
__global__ void chamfer_final_kernel(const float* __restrict__ colmin_ws,
                                     const float* __restrict__ rowsum_ws,
                                     float* __restrict__ out) {
  __shared__ float part[256];
  const int b = blockIdx.x;
  float s = 0.0f;
  for (int m = threadIdx.x; m < NPTS; m += 256) s += colmin_ws[(size_t)b * NPTS + m];
  part[threadIdx.x] = s;
  __syncthreads();
  for (int off = 128; off > 0; off >>= 1) {
    if (threadIdx.x < off) part[threadIdx.x] += part[threadIdx.x + off];
    __syncthreads();
  }
  if (threadIdx.x == 0) {
    float rs = 0.0f;
#pragma unroll
    for (int g = 0; g < ROWGROUPS; ++g) rs += rowsum_ws[b * ROWGROUPS + g];
    out[b] = (part[0] + rs) * (1.0f / NPTS);
  }
}

// ---------------------------------------------------------------------------
extern "C" void kernel_launch(void* const* d_in, const int* in_sizes, int n_in,
                              void* d_out, int out_size, void* d_ws, size_t ws_size,
                              hipStream_t stream) {
  const float* src   = (const float*)d_in[0];  // [4,16,2048,3]
  const float* tgt   = (const float*)d_in[1];  // [16,2048,3]
  const float* rot   = (const float*)d_in[2];  // [4,16,3]
  const float* trans = (const float*)d_in[3];  // [4,16,3]
  const float* scl   = (const float*)d_in[4];  // [4,16]
  float* out = (float*)d_out;                  // [64]

  float* colmin_ws = (float*)d_ws;                       // NBATCH*NPTS floats
  float* rowsum_ws = colmin_ws + (size_t)NBATCH * NPTS;  // NBATCH*ROWGROUPS floats

  const int init_n = NBATCH * NPTS;
  chamfer_init_kernel<<<(init_n + 255) / 256, 256, 0, stream>>>(colmin_ws, rowsum_ws);

  dim3 grid(NBATCH, ROWGROUPS);
  chamfer_wmma_kernel<<<grid, THREADS, 0, stream>>>(src, tgt, rot, trans, scl,
                                                    colmin_ws, rowsum_ws);

  chamfer_final_kernel<<<NBATCH, 256, 0, stream>>>(colmin_ws, rowsum_ws, out);
}